// Light_UI_layer_79774722556259
// MI455X (gfx1250) — compile-verified
//
#include <hip/hip_runtime.h>

typedef __attribute__((ext_vector_type(2))) float v2f;
typedef __attribute__((ext_vector_type(8))) float v8f;

#define NEG_SLOPE 0.2f

// ---------------------------------------------------------------------------
// Zero the workspace accumulators (float4 grid-stride).
// ---------------------------------------------------------------------------
__global__ void lgcn_zero(float* __restrict__ ws, long n) {
    long n4 = n >> 2;                       // n is a multiple of 4
    float4 z = make_float4(0.f, 0.f, 0.f, 0.f);
    long stride = (long)gridDim.x * blockDim.x;
    for (long i = (long)blockIdx.x * blockDim.x + threadIdx.x; i < n4; i += stride)
        ((float4*)ws)[i] = z;
}

// ---------------------------------------------------------------------------
// SpMM scatter: out[row[e], :] += val[e] * emb[col[e], :]
// One wave32 per edge; lane handles dims {lane, lane+32}. Gathers are
// coalesced 128B bursts per half-row; scatters use HW fp32 atomics
// (tables are L2-resident on MI455X: 25.6/51.2 MB << 192 MB L2).
// ---------------------------------------------------------------------------
__global__ void lgcn_spmm_scatter(const int* __restrict__ row,
                                  const int* __restrict__ col,
                                  const float* __restrict__ val,
                                  const float* __restrict__ emb,
                                  float* __restrict__ out, int E) {
    int wave = blockIdx.x * (blockDim.x >> 5) + (threadIdx.x >> 5);
    int lane = threadIdx.x & 31;
    if (wave >= E) return;

    int   r = row[wave];
    int   c = col[wave];
    float v = val[wave];

    const float* src = emb + (long)c * 64;
    float*       dst = out + (long)r * 64;

    float m0 = v * src[lane];
    float m1 = v * src[lane + 32];
#if defined(__HIP_DEVICE_COMPILE__)
    unsafeAtomicAdd(&dst[lane],      m0);   // -> global_atomic_add_f32
    unsafeAtomicAdd(&dst[lane + 32], m1);
#else
    atomicAdd(&dst[lane],      m0);
    atomicAdd(&dst[lane + 32], m1);
#endif
}

// ---------------------------------------------------------------------------
// out = leaky_relu(h @ W) with V_WMMA_F32_16X16X4_F32.
// Block = 256 threads (8 waves). W (64x64 fp32) staged in LDS *pre-interleaved
// as K-pairs*:  Wp[k/2][n] = { W[k][n], W[k+1][n] }  (16 KB total), so each
// B fragment is one contiguous ds_load_b64 straight into the WMMA source
// VGPR pair -- no register shuffling in the inner loop.
//
// Each wave owns a 16-row stripe and computes all four 16-col tiles,
// accumulating over K=64 in steps of 4 (16 iterations x 4 wmma).
//
// Fragment layouts (ISA 7.12.2, 32-bit):
//   A 16x4 : lane L<16 -> M=L,    v0=K0,v1=K1 ; lane>=16 -> M=L-16, v0=K2,v1=K3
//   B 4x16 : lane L<16 -> N=L,    v0=K0,v1=K1 ; lane>=16 -> N=L-16, v0=K2,v1=K3
//   C/D    : VGPR r -> row = r + 8*(lane>>4), col = lane&15
// ---------------------------------------------------------------------------
__device__ __forceinline__ float lrelu(float x) {
    return x > 0.f ? x : NEG_SLOPE * x;
}

__global__ void lgcn_gemm_lrelu_wmma(const float* __restrict__ h,
                                     const float* __restrict__ W,
                                     float* __restrict__ out, int numTiles) {
    __shared__ v2f Wp[32 * 64];   // [k/2][n] -> {W[k][n], W[k+1][n]}

    // Cooperative interleaving stage: 2048 float2 across 256 threads.
    for (int i = threadIdx.x; i < 2048; i += blockDim.x) {
        int kp = i >> 6;          // K-pair index (0..31)
        int n  = i & 63;          // column
        v2f p = { W[(2 * kp) * 64 + n], W[(2 * kp + 1) * 64 + n] };
        Wp[i] = p;
    }
    __syncthreads();

    int wave = threadIdx.x >> 5;
    int lane = threadIdx.x & 31;
    int tile = blockIdx.x * (blockDim.x >> 5) + wave;
    if (tile >= numTiles) return;   // wave-uniform: EXEC stays all-1s in WMMA path

    const int  m     = lane & 15;            // row-within-tile (A) / col (B)
    const int  khalf = (lane >> 4) << 1;     // 0 or 2
    const long rowbase = (long)tile * 16;
    const float* hrow = h + (rowbase + m) * 64;

    v8f acc0 = {}, acc1 = {}, acc2 = {}, acc3 = {};

#pragma unroll
    for (int k0 = 0; k0 < 64; k0 += 4) {
        // A fragment: contiguous float2 at h[row][k0 + khalf]
        v2f a = *(const v2f*)(hrow + k0 + khalf);

        // B fragments: one b64 LDS load per column tile (2addr-fusable)
        const int kp = (k0 >> 1) + (lane >> 4);     // == (k0 + khalf) / 2
        const v2f* wrow = &Wp[kp * 64 + m];
        v2f b0 = wrow[0];
        v2f b1 = wrow[16];
        v2f b2 = wrow[32];
        v2f b3 = wrow[48];

        acc0 = __builtin_amdgcn_wmma_f32_16x16x4_f32(false, a, false, b0, (short)0, acc0, false, false);
        acc1 = __builtin_amdgcn_wmma_f32_16x16x4_f32(false, a, false, b1, (short)0, acc1, false, false);
        acc2 = __builtin_amdgcn_wmma_f32_16x16x4_f32(false, a, false, b2, (short)0, acc2, false, false);
        acc3 = __builtin_amdgcn_wmma_f32_16x16x4_f32(false, a, false, b3, (short)0, acc3, false, false);
    }

    // Leaky-ReLU + store D fragments.
    const int mrow = (lane >> 4) << 3;       // 0 or 8
    const int ncol = lane & 15;
#pragma unroll
    for (int r = 0; r < 8; ++r) {
        float* orow = out + (rowbase + mrow + r) * 64;
        orow[ncol]      = lrelu(acc0[r]);
        orow[ncol + 16] = lrelu(acc1[r]);
        orow[ncol + 32] = lrelu(acc2[r]);
        orow[ncol + 48] = lrelu(acc3[r]);
    }
}

// ---------------------------------------------------------------------------
// Host-side launcher. Inputs (setup_inputs order):
//   0 users_emb [N_U,64] f32   1 items_emb [N_I,64] f32
//   2 u_row i32  3 u_col i32  4 u_val f32
//   5 i_row i32  6 i_col i32  7 i_val f32
//   8 W_u [64,64] f32          9 W_i [64,64] f32
// Output: h_u [N_U,64] f32 then h_i [N_I,64] f32 concatenated.
// ---------------------------------------------------------------------------
extern "C" void kernel_launch(void* const* d_in, const int* in_sizes, int n_in,
                              void* d_out, int out_size, void* d_ws, size_t ws_size,
                              hipStream_t stream) {
    const float* users_emb = (const float*)d_in[0];
    const float* items_emb = (const float*)d_in[1];
    const int*   u_row     = (const int*)d_in[2];
    const int*   u_col     = (const int*)d_in[3];
    const float* u_val     = (const float*)d_in[4];
    const int*   i_row     = (const int*)d_in[5];
    const int*   i_col     = (const int*)d_in[6];
    const float* i_val     = (const float*)d_in[7];
    const float* W_u       = (const float*)d_in[8];
    const float* W_i       = (const float*)d_in[9];

    const int D   = 64;
    const int N_U = in_sizes[0] / D;
    const int N_I = in_sizes[1] / D;
    const int E_U = in_sizes[2];
    const int E_I = in_sizes[5];

    float* h_u = (float*)d_ws;
    float* h_i = h_u + (size_t)N_U * D;
    float* o_u = (float*)d_out;
    float* o_i = o_u + (size_t)N_U * D;

    // 1) zero workspace accumulators (re-done every call: deterministic)
    long totalF = (long)(N_U + N_I) * D;
    lgcn_zero<<<2048, 256, 0, stream>>>((float*)d_ws, totalF);

    // 2) SpMM scatter: one wave per edge, 8 waves per 256-thread block
    lgcn_spmm_scatter<<<(E_U + 7) / 8, 256, 0, stream>>>(u_row, u_col, u_val, users_emb, h_u, E_U);
    lgcn_spmm_scatter<<<(E_I + 7) / 8, 256, 0, stream>>>(i_row, i_col, i_val, items_emb, h_i, E_I);

    // 3) fused GEMM + leaky-ReLU via WMMA (N_U, N_I are multiples of 16)
    int tU = (N_U + 15) / 16;
    int tI = (N_I + 15) / 16;
    lgcn_gemm_lrelu_wmma<<<(tU + 7) / 8, 256, 0, stream>>>(h_u, W_u, o_u, tU);
    lgcn_gemm_lrelu_wmma<<<(tI + 7) / 8, 256, 0, stream>>>(h_i, W_i, o_i, tI);
}